// Predict_loss_test_22299470201301
// MI455X (gfx1250) — compile-verified
//
#include <hip/hip_runtime.h>
#include <hip/hip_bf16.h>

#define MAX_GROUP_MEMBER 10

typedef float v4f __attribute__((ext_vector_type(4)));

// Consume one float4 (indices base..base+3) in increasing-index order so the
// first-max-index tie-break of jnp.argmax is preserved.
__device__ __forceinline__ void amax4(const v4f x, int base, float& best, int& bidx)
{
    if (x.x > best) { best = x.x; bidx = base;     }
    if (x.y > best) { best = x.y; bidx = base + 1; }
    if (x.z > best) { best = x.z; bidx = base + 2; }
    if (x.w > best) { best = x.w; bidx = base + 3; }
}

// ---------------------------------------------------------------------------
// Row-wise argmax (first-index tie-break, matching jnp.argmax).
// One 256-thread block (8 wave32s) per row. Main stream is unrolled x4 so each
// wave keeps four independent 512B non-temporal b128 loads in flight
// (s_wait_loadcnt 3/2/1/0 pattern) instead of one; the tail is software-
// pipelined (issue next load before consuming current). Streaming data is
// read-once => TH_LOAD_NT keeps the 393MB stream from thrashing the 192MB L2.
// ---------------------------------------------------------------------------
__global__ __launch_bounds__(256) void argmax_rows_kernel(
    const float* __restrict__ mat, int M, int* __restrict__ out_idx)
{
    const int row = blockIdx.x;
    const int tid = threadIdx.x;
    const v4f* __restrict__ pv = (const v4f*)(mat + (size_t)row * (size_t)M);
    const int nvec = M >> 2;          // M is a multiple of 4 (4000 / 20000)

    float best = -__builtin_inff();
    int   bidx = 0x7fffffff;

    // ---- main stream: chunks of 1024 vectors (16 KB), 4 loads in flight ----
    const int CH      = 256 * 4;
    const int nchunks = nvec / CH;
    for (int c = 0; c < nchunks; ++c) {
        const int b = c * CH + tid;
        // one speculative line-prefetch per chunk (gfx1250 global_prefetch_b8)
        __builtin_prefetch(&pv[b + 2 * CH], 0, 0);
        v4f x0 = __builtin_nontemporal_load(&pv[b]);
        v4f x1 = __builtin_nontemporal_load(&pv[b + 256]);
        v4f x2 = __builtin_nontemporal_load(&pv[b + 512]);
        v4f x3 = __builtin_nontemporal_load(&pv[b + 768]);
        amax4(x0, (b)       << 2, best, bidx);
        amax4(x1, (b + 256) << 2, best, bidx);
        amax4(x2, (b + 512) << 2, best, bidx);
        amax4(x3, (b + 768) << 2, best, bidx);
    }

    // ---- software-pipelined tail (covers all of M_REC=4000 rows too) ----
    int v = nchunks * CH + tid;
    if (v < nvec) {
        v4f cur = __builtin_nontemporal_load(&pv[v]);
        int vi  = v;
        for (v += 256; v < nvec; v += 256) {
            v4f nxt = __builtin_nontemporal_load(&pv[v]);   // in flight while
            amax4(cur, vi << 2, best, bidx);                // consuming cur
            cur = nxt;
            vi  = v;
        }
        amax4(cur, vi << 2, best, bidx);
    }

    // ---- cross-lane / cross-wave reduction via LDS tree ----
    __shared__ float s_val[256];
    __shared__ int   s_idx[256];
    s_val[tid] = best;
    s_idx[tid] = bidx;
    __syncthreads();

    #pragma unroll
    for (int s = 128; s > 0; s >>= 1) {
        if (tid < s) {
            const float ov = s_val[tid + s];
            const int   oi = s_idx[tid + s];
            if (ov > s_val[tid] || (ov == s_val[tid] && oi < s_idx[tid])) {
                s_val[tid] = ov;
                s_idx[tid] = oi;
            }
        }
        __syncthreads();
    }
    if (tid == 0) out_idx[row] = s_idx[0];
}

// ---------------------------------------------------------------------------
// Per-row gathers: p_self, Thaptic, Tprefer.  O(N) loads, negligible.
// Output layout: out[0..N) = Tprefer, out[N..2N) = Tsocial, out[2N..3N) = Thaptic
// ---------------------------------------------------------------------------
__global__ __launch_bounds__(256) void gather_kernel(
    const int* __restrict__ ri, const int* __restrict__ si,
    const float* __restrict__ preference,   // (P, N)
    const float* __restrict__ structure,    // (M_SUB, M_REC)
    const int* __restrict__ vid,            // (MAX_GROUP_MEMBER + M_REC)
    float* __restrict__ out, float* __restrict__ p_self,
    int N, int M_rec)
{
    const int i = blockIdx.x * blockDim.x + threadIdx.x;
    if (i >= N) return;

    const int r = ri[i];
    const int s = si[i];

    const float ps = preference[(size_t)s * (size_t)N + i];
    p_self[i] = ps;

    // Thaptic = structure[si, ri]
    out[2 * N + i] = structure[(size_t)s * (size_t)M_rec + r];

    // Tprefer = p_self - preference[Vid[MAX_GROUP_MEMBER + ri], users]
    const int v = vid[MAX_GROUP_MEMBER + r];
    out[i] = ps - preference[(size_t)v * (size_t)N + i];
}

// ---------------------------------------------------------------------------
// Tsocial: cache full si[] and p_self[] vectors in LDS (32 KB of the 320 KB
// WGP LDS), then each thread scans all N partners out of LDS. 16 blocks total.
// ---------------------------------------------------------------------------
__global__ __launch_bounds__(256) void tsocial_kernel(
    const int* __restrict__ si, const float* __restrict__ p_self,
    float* __restrict__ out, int N)
{
    __shared__ int   s_si[4096];
    __shared__ float s_ps[4096];

    for (int k = threadIdx.x; k < N; k += blockDim.x) {
        s_si[k] = si[k];
        s_ps[k] = p_self[k];
    }
    __syncthreads();

    const int i = blockIdx.x * blockDim.x + threadIdx.x;
    if (i >= N) return;

    const int   myS = s_si[i];
    const float myP = s_ps[i];
    float acc = 0.0f;

    #pragma unroll 4
    for (int j = 0; j < N; ++j) {
        const float m = fminf(myP, s_ps[j]);
        const bool take = (s_si[j] == myS) & (j != i);
        acc += take ? m : 0.0f;
    }
    out[N + i] = acc;
}

// ---------------------------------------------------------------------------
// Host-side launcher.
// Input order: 0 Recommended_m, 1 Substitute_m, 2 ItemGroups_m(unused),
//              3 Vid, 4 VUU, 5 KUU, 6 Vscore, 7 Kscore, 8 preference, 9 structure
// ---------------------------------------------------------------------------
extern "C" void kernel_launch(void* const* d_in, const int* in_sizes, int n_in,
                              void* d_out, int out_size, void* d_ws, size_t ws_size,
                              hipStream_t stream)
{
    const float* Rec       = (const float*)d_in[0];
    const float* Sub       = (const float*)d_in[1];
    const int*   Vid       = (const int*)  d_in[3];
    const float* pref      = (const float*)d_in[8];
    const float* structure = (const float*)d_in[9];
    float*       out       = (float*)d_out;

    const int N     = in_sizes[4];           // 4096 (len of VUU)
    const int M_rec = in_sizes[0] / N;       // 4000
    const int M_sub = in_sizes[1] / N;       // 20000

    // workspace: ri[N] ints, si[N] ints, p_self[N] floats
    int*   ri     = (int*)d_ws;
    int*   si     = ri + N;
    float* p_self = (float*)(si + N);

    argmax_rows_kernel<<<N, 256, 0, stream>>>(Rec, M_rec, ri);
    argmax_rows_kernel<<<N, 256, 0, stream>>>(Sub, M_sub, si);

    const int blocks = (N + 255) / 256;
    gather_kernel<<<blocks, 256, 0, stream>>>(ri, si, pref, structure, Vid,
                                              out, p_self, N, M_rec);
    tsocial_kernel<<<blocks, 256, 0, stream>>>(si, p_self, out, N);
}